// TopBatchDrop_944892805646
// MI455X (gfx1250) — compile-verified
//
#include <hip/hip_runtime.h>
#include <hip/hip_bf16.h>

// Problem constants (from reference): x is (64, 2048, 24, 12) f32
#define B_  64
#define C_  2048
#define H_  24
#define W_  12
#define HW_ (H_ * W_)      // 288
#define RH_ 8              // round(0.33 * 24)

typedef float v2f __attribute__((ext_vector_type(2)));
typedef float v8f __attribute__((ext_vector_type(8)));

// ---------------------------------------------------------------------------
// Phase 1: act[b, s] = sum_c x[b, c, s]^2   via V_WMMA_F32_16X16X4_F32.
// One wave per (sample, 16-position group). D = A x Ones + C: every column of
// the 16x16 accumulator holds the running channel-sum for its row's position.
// A-matrix 16x4 f32 layout (ISA 7.12.2): VGPR0: lanes0-15 K=0, lanes16-31 K=2;
// VGPR1: K=1 / K=3; row M = lane & 15.  B = all-ones (layout-independent).
// ---------------------------------------------------------------------------
__global__ __launch_bounds__(32)
void energy_wmma_kernel(const float* __restrict__ x, float* __restrict__ act) {
    const int task = blockIdx.x;          // 0 .. 64*18-1
    const int b    = task / (HW_ / 16);   // sample
    const int g    = task % (HW_ / 16);   // 16-position group
    const int lane = threadIdx.x;         // wave32
    const int M    = lane & 15;           // A-matrix row
    const int K0   = (lane >> 4) << 1;    // 0 or 2 (VGPR0 K; VGPR1 holds K0+1)

    const float* base = x + (size_t)b * C_ * HW_ + g * 16 + M;

    v8f acc = {};
    v2f ones; ones.x = 1.0f; ones.y = 1.0f;

    for (int c0 = 0; c0 < C_; c0 += 4) {
        float a0 = base[(size_t)(c0 + K0)     * HW_];
        float a1 = base[(size_t)(c0 + K0 + 1) * HW_];
        v2f A; A.x = a0 * a0; A.y = a1 * a1;
        // D = A(16x4) x Ones(4x16) + C  -> row sums broadcast across columns
        acc = __builtin_amdgcn_wmma_f32_16x16x4_f32(
            /*neg_a=*/false, A, /*neg_b=*/false, ones,
            /*c_mod=*/(short)0, acc, /*reuse_a=*/false, /*reuse_b=*/false);
    }

    // C/D layout: VGPR r, lanes 0-15 -> (M=r, N=lane); lanes 16-31 -> (M=8+r).
    // Column N=0 lives in lanes 0 (rows 0-7) and 16 (rows 8-15).
    if ((lane & 15) == 0) {
        const int half = lane >> 4;
        float* dst = act + (size_t)b * HW_ + g * 16 + half * 8;
#pragma unroll
        for (int r = 0; r < 8; ++r) dst[r] = acc[r];
    }
}

// ---------------------------------------------------------------------------
// Phase 2: per sample, row-max over w, drop the 8 largest rows (argsort-stable
// tie-break: among equal maxima, larger index is dropped first).
// One wave per sample.
// ---------------------------------------------------------------------------
__global__ __launch_bounds__(32)
void select_mask_kernel(const float* __restrict__ act, float* __restrict__ mask) {
    const int b    = blockIdx.x;
    const int lane = threadIdx.x;
    __shared__ float rowmax[H_];

    if (lane < H_) {
        const float* a = act + (size_t)b * HW_ + lane * W_;
        float m = a[0];
#pragma unroll
        for (int j = 1; j < W_; ++j) m = fmaxf(m, a[j]);
        rowmax[lane] = m;
    }
    __syncthreads();

    if (lane < H_) {
        const float mr = rowmax[lane];
        int greater = 0;
#pragma unroll
        for (int j = 0; j < H_; ++j) {
            float mj = rowmax[j];
            greater += (mj > mr) || (mj == mr && j > lane);
        }
        mask[b * H_ + lane] = (greater < RH_) ? 0.0f : 1.0f;
    }
}

// ---------------------------------------------------------------------------
// Phase 3: out = x * mask[b, h], streamed as float4 (w=12 -> each float4 lies
// inside one row h, so one mask scalar per vector). Pure bandwidth kernel.
// ---------------------------------------------------------------------------
__global__ __launch_bounds__(256)
void apply_mask_kernel(const float* __restrict__ x, const float* __restrict__ mask,
                       float* __restrict__ out) {
    const int HW4 = HW_ / 4;   // 72 float4 per (b,c) plane
    const int W4  = W_ / 4;    // 3 float4 per row

    size_t i4  = (size_t)blockIdx.x * blockDim.x + threadIdx.x;
    size_t row = i4 / HW4;          // b*C_ + c
    int    q   = (int)(i4 - row * HW4);
    int    h   = q / W4;
    int    b   = (int)(row / C_);

    const float m = mask[b * H_ + h];
    float4 v = ((const float4*)x)[i4];
    v.x *= m; v.y *= m; v.z *= m; v.w *= m;
    ((float4*)out)[i4] = v;
}

// ---------------------------------------------------------------------------
extern "C" void kernel_launch(void* const* d_in, const int* in_sizes, int n_in,
                              void* d_out, int out_size, void* d_ws, size_t ws_size,
                              hipStream_t stream) {
    (void)in_sizes; (void)n_in; (void)out_size; (void)ws_size;

    const float* x   = (const float*)d_in[0];
    float*       out = (float*)d_out;

    // workspace: act (64*288 f32) then mask (64*24 f32); fully overwritten
    // before being read each call, so 0xAA poison is irrelevant.
    float* act  = (float*)d_ws;
    float* mask = act + (size_t)B_ * HW_;

    // Phase 1: 64 samples x 18 sixteen-position groups, one wave each.
    energy_wmma_kernel<<<dim3(B_ * (HW_ / 16)), dim3(32), 0, stream>>>(x, act);

    // Phase 2: one wave per sample.
    select_mask_kernel<<<dim3(B_), dim3(32), 0, stream>>>(act, mask);

    // Phase 3: 37,748,736 elements / 4 = 9,437,184 float4s; 256 threads/block.
    const size_t n4 = (size_t)B_ * C_ * HW_ / 4;           // 9437184
    apply_mask_kernel<<<dim3((unsigned)(n4 / 256)), dim3(256), 0, stream>>>(x, mask, out);
}